// UpSamp_36464272343205
// MI455X (gfx1250) — compile-verified
//
#include <hip/hip_runtime.h>
#include <stdint.h>

// Problem constants (LVL = 7)
#define BC_N     1024      // B*C = 16*64
#define NV_PREV  40962     // 10*4^6 + 2
#define NV_UP    163842    // 10*4^7 + 2
#define NV_PAD   122880    // NV_UP - NV_PREV
#define NF_PREV  81920     // 20*4^6

// ---------------------------------------------------------------------------
// Pass 0: initialize inverse map to "untouched" (-1 => pad value 1.0f)
// ---------------------------------------------------------------------------
__global__ void init_inv_kernel(int2* __restrict__ inv) {
    int p = blockIdx.x * blockDim.x + threadIdx.x;
    if (p < NV_PAD) inv[p] = make_int2(-1, -1);
}

// ---------------------------------------------------------------------------
// Passes 1..3: build inverse map. Each pass scatters (s0,s1) source indices
// into inv[] at p = F_mid[f,k] - NV_PREV. Within one pass the p's are unique
// (F_mid columns are permutations); the three passes are launched
// sequentially on the stream, so later passes overwrite earlier ones exactly
// like the reference's sequential .at[].set calls.
// ---------------------------------------------------------------------------
__global__ void scatter_pass_kernel(const int* __restrict__ F_prev,
                                    const int* __restrict__ F_mid,
                                    int2* __restrict__ inv,
                                    int a, int b, int k) {
    int f = blockIdx.x * blockDim.x + threadIdx.x;
    if (f >= NF_PREV) return;
    int s0 = F_prev[f * 3 + a];
    int s1 = F_prev[f * 3 + b];
    int p  = F_mid [f * 3 + k] - NV_PREV;
    inv[p] = make_int2(s0, s1);
}

// ---------------------------------------------------------------------------
// Main kernel: one workgroup per (b,c) slice.
//  1) Stage the whole 164KB x-slice into LDS (CDNA5: 320KB LDS per WGP)
//     via async global->LDS b64 copies (ASYNCcnt path).
//  2) Stream out the copy region from LDS with coalesced NT b64 stores.
//  3) Fill the pad region: coalesced NT b64 stores, operands gathered from
//     LDS; inv stays L2-resident because the output stream is non-temporal.
// ---------------------------------------------------------------------------
#if defined(__gfx1250__) && __has_builtin(__builtin_amdgcn_global_load_async_to_lds_b64)
#define USE_ASYNC_LDS 1
typedef int v2i __attribute__((vector_size(2 * sizeof(int))));
typedef __attribute__((address_space(1))) v2i* g_v2i_ptr;   // global
typedef __attribute__((address_space(3))) v2i* l_v2i_ptr;   // LDS
#endif

typedef float f2v __attribute__((ext_vector_type(2)));

__global__ __launch_bounds__(1024)
void upsample_kernel(const float* __restrict__ x,
                     const int2*  __restrict__ inv,
                     float*       __restrict__ out) {
    extern __shared__ float lds[];   // NV_PREV floats = 163848 B dynamic LDS

    const int    bc   = blockIdx.x;
    const float* xs   = x   + (size_t)bc * NV_PREV;
    float*       outs = out + (size_t)bc * NV_UP;
    const int    tid  = threadIdx.x;
    const int    nth  = blockDim.x;

    // --- Stage slice into LDS (slice base is 8B-aligned -> use b64) ---------
#ifdef USE_ASYNC_LDS
    for (int i = tid; i < NV_PREV / 2; i += nth) {           // 20481 float2's
        __builtin_amdgcn_global_load_async_to_lds_b64(
            (g_v2i_ptr)(uintptr_t)(xs + 2 * i),
            (l_v2i_ptr)(uint32_t)(uintptr_t)(lds + 2 * i),
            /*offset=*/0, /*cpol=*/0);
    }
#if __has_builtin(__builtin_amdgcn_s_wait_asynccnt)
    __builtin_amdgcn_s_wait_asynccnt(0);
#else
    asm volatile("s_wait_asynccnt 0" ::: "memory");
#endif
#else
    for (int i = tid; i < NV_PREV / 2; i += nth) {
        f2v v = *(const f2v*)(xs + 2 * i);
        *(f2v*)(lds + 2 * i) = v;
    }
#endif
    __syncthreads();

    // --- Copy region: LDS -> global, coalesced non-temporal b64 stores ------
    for (int i = tid; i < NV_PREV / 2; i += nth) {
        f2v v = *(const f2v*)(lds + 2 * i);
        __builtin_nontemporal_store(v, (f2v*)(outs + 2 * i));
    }

    // --- Pad region: gather midpoints from LDS, coalesced NT b64 stores -----
    // inv is ~1MB shared across all 1024 blocks -> stays L2-resident since
    // the output stream bypasses cache retention (NT policy).
    const int4* inv4 = (const int4*)inv;                     // 2 entries / load
    float* pout = outs + NV_PREV;                            // 8B-aligned
    for (int q = tid; q < NV_PAD / 2; q += nth) {            // 61440 pairs
        int4 e = inv4[q];
        f2v v;
        v.x = (e.x >= 0) ? 0.5f * (lds[e.x] + lds[e.y]) : 1.0f;
        v.y = (e.z >= 0) ? 0.5f * (lds[e.z] + lds[e.w]) : 1.0f;
        __builtin_nontemporal_store(v, (f2v*)(pout + 2 * q));
    }
}

// ---------------------------------------------------------------------------
extern "C" void kernel_launch(void* const* d_in, const int* in_sizes, int n_in,
                              void* d_out, int out_size, void* d_ws, size_t ws_size,
                              hipStream_t stream) {
    const float* x      = (const float*)d_in[0];
    const int*   F_prev = (const int*)  d_in[1];
    const int*   F_mid  = (const int*)  d_in[2];
    float*       out    = (float*)d_out;
    int2*        inv    = (int2*)d_ws;   // NV_PAD * 8 B = 983040 B of scratch

    // Build inverse map (ordered passes replicate sequential .at[].set).
    init_inv_kernel<<<(NV_PAD + 255) / 256, 256, 0, stream>>>(inv);
    scatter_pass_kernel<<<(NF_PREV + 255) / 256, 256, 0, stream>>>(
        F_prev, F_mid, inv, 0, 1, 0);   // F_mid[:,0] = (v0+v1)/2
    scatter_pass_kernel<<<(NF_PREV + 255) / 256, 256, 0, stream>>>(
        F_prev, F_mid, inv, 1, 2, 1);   // F_mid[:,1] = (v1+v2)/2
    scatter_pass_kernel<<<(NF_PREV + 255) / 256, 256, 0, stream>>>(
        F_prev, F_mid, inv, 2, 1, 2);   // F_mid[:,2] = (v2+v1)/2

    // One workgroup per (b,c) slice; 32 waves; 164KB dynamic LDS per WGP.
    upsample_kernel<<<BC_N, 1024, NV_PREV * sizeof(float), stream>>>(x, inv, out);
}